// CGLayer_36163624633140
// MI455X (gfx1250) — compile-verified
//
#include <hip/hip_runtime.h>
#include <hip/hip_bf16.h>
#include <math.h>

// ---------------------------------------------------------------------------
// Clifford GNN layer forward for gfx1250 (MI455X).
// Dense h-side MLPs run on V_WMMA_F32_16X16X4_F32 (fp32-exact, 16x16 tiles,
// K stepped by 4). Geometric-algebra parts (mv_linear + weighted geometric
// product, K=51/34) run on VALU with LDS-staged multivectors. BatchNorm and
// segment-mean use fp32 global atomics.
// ---------------------------------------------------------------------------

#define NN 10000      // nodes
#define NE 100000     // edges
#define HID 128
#define K_PHI 448     // m_inv(64) + h_i(128) + h_j(128) + (h_i-h_j)(128)
#define K_TH_REAL 322 // h(128) + h_red(128) + u_inv(64) + node_attr_h(2)
#define K_TH 324      // padded to multiple of 4
#define FIN_PHI 51    // x_i(16)+x_j(16)+(x_i-x_j)(16)+edge_attr_x(3)
#define FIN_TH 34     // x(16)+x_red(16)+node_attr_x(2)

typedef float v2f __attribute__((ext_vector_type(2)));
typedef float v8f __attribute__((ext_vector_type(8)));

// Cl(3,0) blade order: 1, e1, e2, e3, e12, e13, e23, e123
static constexpr int   kGrade[8]  = {0,1,1,1,2,2,2,3};
static constexpr int   kProdK[64] = {
  0,1,2,3,4,5,6,7,
  1,0,4,5,2,3,7,6,
  2,4,0,6,1,7,3,5,
  3,5,6,0,7,1,2,4,
  4,2,1,7,0,6,5,3,
  5,3,7,1,6,0,4,2,
  6,7,3,2,5,4,0,1,
  7,6,5,4,3,2,1,0 };
static constexpr float kProdS[64] = {
   1, 1, 1, 1, 1, 1, 1, 1,
   1, 1, 1, 1, 1, 1, 1, 1,
   1,-1, 1, 1,-1,-1, 1,-1,
   1,-1,-1, 1, 1,-1,-1, 1,
   1,-1, 1, 1,-1,-1, 1,-1,
   1,-1,-1, 1, 1,-1,-1, 1,
   1, 1,-1, 1,-1, 1,-1,-1,
   1, 1,-1, 1,-1, 1,-1,-1 };

__device__ inline void atomAddF(float* p, float v) {
  __hip_atomic_fetch_add(p, v, __ATOMIC_RELAXED, __HIP_MEMORY_SCOPE_AGENT);
}

// -------------------------- WMMA inner loop --------------------------------
// Tile: 16 rows (edges/nodes) x 16 cols (features). A staged in LDS row-major
// (16 x K). B is weight matrix row-major (out x K); B[k][n] = W[n*K + k].
// A layout (ISA 7.12.2, 32-bit A 16x4): lanes 0-15 hold K=kk+0/kk+1 in the
// two A VGPRs, lanes 16-31 hold K=kk+2/kk+3. C/D: VGPR r = row r (lanes 0-15)
// / row r+8 (lanes 16-31), col = lane&15.
__device__ inline v8f wmma_k_loop(const float* __restrict__ sA, int K,
                                  const float* __restrict__ W, int nglob,
                                  int lane, v8f acc)
{
  const int m    = lane & 15;
  const int koff = (lane >> 4) << 1;   // 0 or 2
  const float* wrow = W + (size_t)nglob * K;
  for (int kk = 0; kk < K; kk += 4) {
    if ((kk & 60) == 0) __builtin_prefetch(wrow + kk + 64, 0, 1);
    v2f a, b;
    a.x = sA[m * K + kk + koff];
    a.y = sA[m * K + kk + koff + 1];
    b.x = wrow[kk + koff];
    b.y = wrow[kk + koff + 1];
    acc = __builtin_amdgcn_wmma_f32_16x16x4_f32(false, a, false, b,
                                                (short)0, acc, false, false);
  }
  return acc;
}

__device__ inline void store_rows(float* __restrict__ dst, int rowbase,
                                  int nglob, int lane, v8f acc, bool relu)
{
  const int r0 = rowbase + ((lane >> 4) << 3);
  #pragma unroll
  for (int r = 0; r < 8; ++r) {
    float v = acc[r];
    if (relu) v = fmaxf(v, 0.f);
    dst[(size_t)(r0 + r) * HID + nglob] = v;
  }
}

// -------------------------- utility kernels --------------------------------
__global__ void zero_kernel(float* p, long n) {
  long i = (long)blockIdx.x * blockDim.x + threadIdx.x;
  if (i < n) p[i] = 0.f;
}

__global__ void pad_w1_kernel(const float* __restrict__ W, float* __restrict__ Wp) {
  int t = blockIdx.x * blockDim.x + threadIdx.x;
  if (t >= HID * K_TH) return;
  int o = t / K_TH, k = t % K_TH;
  Wp[t] = (k < K_TH_REAL) ? W[o * K_TH_REAL + k] : 0.f;
}

// ---------------------- geometric product per channel ----------------------
// out[i] = sum_n cat[n][i]*Wl[m,n,g(i)]  (+bl at i=0)
//        + sum_n sum_{a,b} cat[n][a]*y[n][b]*Cay[a,b,k]*Wgp[m,n,g(a),g(b),g(k)]
template<int FIN>
__device__ inline void fcgp_channel(const float* __restrict__ catp,
                                    const float* __restrict__ yp,
                                    const float* __restrict__ Wl,
                                    const float* __restrict__ bl,
                                    const float* __restrict__ Wgp,
                                    int m, float out[8])
{
  #pragma unroll
  for (int i = 0; i < 8; ++i) out[i] = 0.f;
  for (int n = 0; n < FIN; ++n) {
    const float* wl = Wl  + (size_t)(m * FIN + n) * 4;
    const float* wg = Wgp + (size_t)(m * FIN + n) * 64;
    float cv[8], yv[8];
    #pragma unroll
    for (int i = 0; i < 8; ++i) { cv[i] = catp[n*8+i]; yv[i] = yp[n*8+i]; }
    #pragma unroll
    for (int i = 0; i < 8; ++i) out[i] += cv[i] * wl[kGrade[i]];
    #pragma unroll
    for (int a = 0; a < 8; ++a) {
      #pragma unroll
      for (int b = 0; b < 8; ++b) {
        const int k = kProdK[a*8+b];
        out[k] += cv[a] * yv[b] * kProdS[a*8+b]
                * wg[kGrade[a]*16 + kGrade[b]*4 + kGrade[k]];
      }
    }
  }
  out[0] += bl[m];
}

// -------------------------- edge geometry kernel ---------------------------
// 8 edges per block of 128 threads (16 channel-threads per edge).
__global__ __launch_bounds__(128) void edge_geom_kernel(
    const float* __restrict__ x, const int* __restrict__ edges,
    const float* __restrict__ eax,
    const float* __restrict__ Wr, const float* __restrict__ Wl,
    const float* __restrict__ bl, const float* __restrict__ Wgp,
    const float* __restrict__ lna,
    float* __restrict__ m_x, float* __restrict__ m_inv)
{
  __shared__ float s_cat[8 * FIN_PHI * 8];
  __shared__ float s_y  [8 * FIN_PHI * 8];
  __shared__ float s_nrm[8 * 16];
  const int tid = threadIdx.x;
  const int ebase = blockIdx.x * 8;

  for (int t = tid; t < 8 * FIN_PHI * 8; t += 128) {
    int le = t / (FIN_PHI * 8), rem = t % (FIN_PHI * 8);
    int r = rem >> 3, c = rem & 7;
    int e = ebase + le;
    int i = edges[e], j = edges[NE + e];
    float v;
    if (r < 16)      v = x[i*128 + r*8 + c];
    else if (r < 32) v = x[j*128 + (r-16)*8 + c];
    else if (r < 48) v = x[i*128 + (r-32)*8 + c] - x[j*128 + (r-32)*8 + c];
    else             v = eax[(size_t)e*24 + (r-48)*8 + c];
    s_cat[t] = v;
  }
  __syncthreads();

  // y = mv_linear(cat, Wr)
  for (int t = tid; t < 8 * FIN_PHI * 8; t += 128) {
    int le = t / (FIN_PHI * 8), rem = t % (FIN_PHI * 8);
    int mm = rem >> 3, i = rem & 7;
    const float* catp = &s_cat[le * FIN_PHI * 8];
    const int g = kGrade[i];
    float acc = 0.f;
    for (int n = 0; n < FIN_PHI; ++n)
      acc += catp[n*8+i] * Wr[(size_t)(mm * FIN_PHI + n) * 4 + g];
    s_y[t] = acc;
  }
  __syncthreads();

  const int le = tid >> 4, m = tid & 15;
  const float* catp = &s_cat[le * FIN_PHI * 8];
  const float* yp   = &s_y  [le * FIN_PHI * 8];
  float out[8];
  fcgp_channel<FIN_PHI>(catp, yp, Wl, bl, Wgp, m, out);
  const float invs2 = 0.70710678118654752f;
  float nr = 0.f;
  #pragma unroll
  for (int i = 0; i < 8; ++i) { out[i] *= invs2; nr += out[i]*out[i]; }
  s_nrm[le*16 + m] = sqrtf(nr);
  __syncthreads();

  float mean = 0.f;
  #pragma unroll
  for (int t = 0; t < 16; ++t) mean += s_nrm[le*16 + t];
  mean = mean * (1.f/16.f) + 1e-6f;
  const float scale = lna[m] / mean;
  const int e = ebase + le;
  float mx[8];
  #pragma unroll
  for (int i = 0; i < 8; ++i) {
    mx[i] = out[i] * scale;
    m_x[(size_t)e*128 + m*8 + i] = mx[i];
  }
  float* iv = &m_inv[(size_t)e*64 + m*4];
  iv[0] = mx[0];
  iv[1] = mx[1]*mx[1] + mx[2]*mx[2] + mx[3]*mx[3];
  iv[2] = mx[4]*mx[4] + mx[5]*mx[5] + mx[6]*mx[6];
  iv[3] = mx[7]*mx[7];
}

// -------------------------- node geometry kernel ---------------------------
__global__ __launch_bounds__(128) void node_geom_kernel(
    const float* __restrict__ x, const float* __restrict__ x_red,
    const float* __restrict__ nax,
    const float* __restrict__ Wr, const float* __restrict__ Wl,
    const float* __restrict__ bl, const float* __restrict__ Wgp,
    const float* __restrict__ lna,
    float* __restrict__ xu_pre, float* __restrict__ u_inv)
{
  __shared__ float s_cat[8 * FIN_TH * 8];
  __shared__ float s_y  [8 * FIN_TH * 8];
  __shared__ float s_nrm[8 * 16];
  const int tid = threadIdx.x;
  const int nbase = blockIdx.x * 8;

  for (int t = tid; t < 8 * FIN_TH * 8; t += 128) {
    int le = t / (FIN_TH * 8), rem = t % (FIN_TH * 8);
    int r = rem >> 3, c = rem & 7;
    int nd = nbase + le;
    float v;
    if (r < 16)      v = x[(size_t)nd*128 + r*8 + c];
    else if (r < 32) v = x_red[(size_t)nd*128 + (r-16)*8 + c];
    else             v = nax[(size_t)nd*16 + (r-32)*8 + c];
    s_cat[t] = v;
  }
  __syncthreads();

  for (int t = tid; t < 8 * FIN_TH * 8; t += 128) {
    int le = t / (FIN_TH * 8), rem = t % (FIN_TH * 8);
    int mm = rem >> 3, i = rem & 7;
    const float* catp = &s_cat[le * FIN_TH * 8];
    const int g = kGrade[i];
    float acc = 0.f;
    for (int n = 0; n < FIN_TH; ++n)
      acc += catp[n*8+i] * Wr[(size_t)(mm * FIN_TH + n) * 4 + g];
    s_y[t] = acc;
  }
  __syncthreads();

  const int le = tid >> 4, m = tid & 15;
  const float* catp = &s_cat[le * FIN_TH * 8];
  const float* yp   = &s_y  [le * FIN_TH * 8];
  float out[8];
  fcgp_channel<FIN_TH>(catp, yp, Wl, bl, Wgp, m, out);
  const float invs2 = 0.70710678118654752f;
  float nr = 0.f;
  #pragma unroll
  for (int i = 0; i < 8; ++i) { out[i] *= invs2; nr += out[i]*out[i]; }
  s_nrm[le*16 + m] = sqrtf(nr);
  __syncthreads();

  float mean = 0.f;
  #pragma unroll
  for (int t = 0; t < 16; ++t) mean += s_nrm[le*16 + t];
  mean = mean * (1.f/16.f) + 1e-6f;
  const float scale = lna[m] / mean;
  const int nd = nbase + le;
  #pragma unroll
  for (int i = 0; i < 8; ++i)
    xu_pre[(size_t)nd*128 + m*8 + i] = out[i] * scale;

  // invariants of ORIGINAL x (reference uses _invariants(x) for theta_h)
  float xv[8];
  #pragma unroll
  for (int i = 0; i < 8; ++i) xv[i] = x[(size_t)nd*128 + m*8 + i];
  float* iv = &u_inv[(size_t)nd*64 + m*4];
  iv[0] = xv[0];
  iv[1] = xv[1]*xv[1] + xv[2]*xv[2] + xv[3]*xv[3];
  iv[2] = xv[4]*xv[4] + xv[5]*xv[5] + xv[6]*xv[6];
  iv[3] = xv[7]*xv[7];
}

// ------------------- GEMM 1: z1 = cat(edge feats) @ W1^T -------------------
__global__ __launch_bounds__(256) void gemm_phi_h1_kernel(
    const float* __restrict__ m_inv, const float* __restrict__ h,
    const int* __restrict__ edges, const float* __restrict__ W1,
    float* __restrict__ z1)
{
  __shared__ float sA[16 * K_PHI];
  const int tid = threadIdx.x;
  const int rowbase = blockIdx.x * 16;
  for (int t = tid; t < 16 * K_PHI; t += 256) {
    int le = t / K_PHI, c = t % K_PHI;
    int e = rowbase + le;
    float v;
    if (c < 64) v = m_inv[(size_t)e*64 + c];
    else {
      int f = (c - 64) & 127, seg = (c - 64) >> 7;
      int i = edges[e], j = edges[NE + e];
      float hi = h[(size_t)i*128 + f], hj = h[(size_t)j*128 + f];
      v = (seg == 0) ? hi : ((seg == 1) ? hj : hi - hj);
    }
    sA[t] = v;
  }
  __syncthreads();
  const int lane = tid & 31, wave = tid >> 5;
  const int nglob = wave * 16 + (lane & 15);
  v8f acc = {0.f,0.f,0.f,0.f,0.f,0.f,0.f,0.f};
  acc = wmma_k_loop(sA, K_PHI, W1, nglob, lane, acc);
  store_rows(z1, rowbase, nglob, lane, acc, false);
}

// ----------------- GEMM: out = act( BN_relu(z) @ W2^T + b2 ) ---------------
template<bool RELU_OUT>
__global__ __launch_bounds__(256) void gemm_bn_w2_kernel(
    const float* __restrict__ z, const float* __restrict__ mu_rstd,
    const float* __restrict__ g, const float* __restrict__ bb,
    const float* __restrict__ W2, const float* __restrict__ b2,
    float* __restrict__ out)
{
  __shared__ float sA[16 * HID];
  const int tid = threadIdx.x;
  const int rowbase = blockIdx.x * 16;
  for (int t = tid; t < 16 * HID; t += 256) {
    int c = t & 127;
    float v = z[(size_t)rowbase * HID + t];
    v = fmaxf(g[c] * (v - mu_rstd[c]) * mu_rstd[128 + c] + bb[c], 0.f);
    sA[t] = v;
  }
  __syncthreads();
  const int lane = tid & 31, wave = tid >> 5;
  const int nglob = wave * 16 + (lane & 15);
  const float bias = b2[nglob];
  v8f acc = {bias,bias,bias,bias,bias,bias,bias,bias};
  acc = wmma_k_loop(sA, HID, W2, nglob, lane, acc);
  store_rows(out, rowbase, nglob, lane, acc, RELU_OUT);
}

// ----------------- GEMM: out = relu( in @ W^T + b ), K=128 -----------------
__global__ __launch_bounds__(256) void gemm_relu_kernel(
    const float* __restrict__ in, const float* __restrict__ W,
    const float* __restrict__ b, float* __restrict__ out)
{
  __shared__ float sA[16 * HID];
  const int tid = threadIdx.x;
  const int rowbase = blockIdx.x * 16;
  for (int t = tid; t < 16 * HID; t += 256)
    sA[t] = in[(size_t)rowbase * HID + t];
  __syncthreads();
  const int lane = tid & 31, wave = tid >> 5;
  const int nglob = wave * 16 + (lane & 15);
  const float bias = b[nglob];
  v8f acc = {bias,bias,bias,bias,bias,bias,bias,bias};
  acc = wmma_k_loop(sA, HID, W, nglob, lane, acc);
  store_rows(out, rowbase, nglob, lane, acc, true);
}

// ------- GEMM gate: w = t @ W2^T + b2 (64 cols); dst = src*sigmoid(w) ------
// Gate col c -> channel m=c>>2, grade gk=c&3, applied to components of grade gk.
__global__ __launch_bounds__(128) void gemm_gate_kernel(
    const float* __restrict__ tin, const float* __restrict__ W2,
    const float* __restrict__ b2, const float* __restrict__ src,
    float* __restrict__ dst)
{
  __shared__ float sA[16 * HID];
  const int tid = threadIdx.x;
  const int rowbase = blockIdx.x * 16;
  for (int t = tid; t < 16 * HID; t += 128)
    sA[t] = tin[(size_t)rowbase * HID + t];
  __syncthreads();
  const int lane = tid & 31, wave = tid >> 5;  // 4 waves -> 64 cols
  const int nglob = wave * 16 + (lane & 15);
  const float bias = b2[nglob];
  v8f acc = {bias,bias,bias,bias,bias,bias,bias,bias};
  acc = wmma_k_loop(sA, HID, W2, nglob, lane, acc);
  const int m = nglob >> 2, gk = nglob & 3;
  const int radd = (lane >> 4) << 3;
  #pragma unroll
  for (int r = 0; r < 8; ++r) {
    const int row = rowbase + radd + r;
    const float sg = 1.f / (1.f + __expf(-acc[r]));
    const float* s = src + (size_t)row*128 + m*8;
    float*       d = dst + (size_t)row*128 + m*8;
    if (gk == 0)      { d[0] = s[0]*sg; }
    else if (gk == 1) { d[1] = s[1]*sg; d[2] = s[2]*sg; d[3] = s[3]*sg; }
    else if (gk == 2) { d[4] = s[4]*sg; d[5] = s[5]*sg; d[6] = s[6]*sg; }
    else              { d[7] = s[7]*sg; }
  }
}

// ------------------- GEMM: z2 = cat(node feats) @ W1p^T + b1 ---------------
__global__ __launch_bounds__(256) void gemm_theta_h1_kernel(
    const float* __restrict__ h, const float* __restrict__ h_red,
    const float* __restrict__ u_inv, const float* __restrict__ nah,
    const float* __restrict__ W1p, const float* __restrict__ b1,
    float* __restrict__ z2)
{
  __shared__ float sA[16 * K_TH];
  const int tid = threadIdx.x;
  const int rowbase = blockIdx.x * 16;
  for (int t = tid; t < 16 * K_TH; t += 256) {
    int le = t / K_TH, c = t % K_TH;
    int nd = rowbase + le;
    float v;
    if (c < 128)      v = h[(size_t)nd*128 + c];
    else if (c < 256) v = h_red[(size_t)nd*128 + (c-128)];
    else if (c < 320) v = u_inv[(size_t)nd*64 + (c-256)];
    else if (c < 322) v = nah[(size_t)nd*2 + (c-320)];
    else              v = 0.f;
    sA[t] = v;
  }
  __syncthreads();
  const int lane = tid & 31, wave = tid >> 5;
  const int nglob = wave * 16 + (lane & 15);
  const float bias = b1[nglob];
  v8f acc = {bias,bias,bias,bias,bias,bias,bias,bias};
  acc = wmma_k_loop(sA, K_TH, W1p, nglob, lane, acc);
  store_rows(z2, rowbase, nglob, lane, acc, false);
}

// -------------------------- batch-norm statistics --------------------------
__global__ __launch_bounds__(128) void bn_stats_kernel(
    const float* __restrict__ z, int rows, float* __restrict__ acc)
{
  const int f = threadIdx.x;
  float s = 0.f, s2 = 0.f;
  for (int r = blockIdx.x; r < rows; r += gridDim.x) {
    float v = z[(size_t)r * HID + f];
    s += v; s2 += v * v;
  }
  atomAddF(&acc[f], s);
  atomAddF(&acc[128 + f], s2);
}

__global__ __launch_bounds__(128) void bn_finalize_kernel(
    const float* __restrict__ acc, float inv_rows, float* __restrict__ out)
{
  const int f = threadIdx.x;
  float mu  = acc[f] * inv_rows;
  float var = acc[128 + f] * inv_rows - mu * mu;
  out[f] = mu;
  out[128 + f] = rsqrtf(var + 1e-5f);
}

// -------------------------- segment mean (scatter) -------------------------
__global__ void scatter_kernel(const float* __restrict__ mx,
                               const float* __restrict__ mh,
                               const int* __restrict__ edges,
                               float* __restrict__ xa, float* __restrict__ ha,
                               float* __restrict__ cnt)
{
  long idx = (long)blockIdx.x * blockDim.x + threadIdx.x;
  const long total = (long)NE * 128;
  const long step = (long)gridDim.x * blockDim.x;
  for (; idx < total; idx += step) {
    int e = (int)(idx >> 7), c = (int)(idx & 127);
    int ni = edges[e];
    atomAddF(&xa[(size_t)ni*128 + c], mx[idx]);
    atomAddF(&ha[(size_t)ni*128 + c], mh[idx]);
    if (c == 0) atomAddF(&cnt[ni], 1.f);
  }
}

__global__ void seg_div_kernel(float* __restrict__ xa, float* __restrict__ ha,
                               const float* __restrict__ cnt)
{
  int idx = blockIdx.x * 256 + threadIdx.x;
  if (idx >= NN * 128) return;
  float inv = 1.f / fmaxf(cnt[idx >> 7], 1.f);
  xa[idx] *= inv;
  ha[idx] *= inv;
}

// ----------------------------------------------------------------------------
extern "C" void kernel_launch(void* const* d_in, const int* in_sizes, int n_in,
                              void* d_out, int out_size, void* d_ws, size_t ws_size,
                              hipStream_t stream) {
  (void)in_sizes; (void)n_in; (void)out_size; (void)ws_size;
  // Inputs in setup_inputs() dict order, params pytree flattened in insertion
  // order (nested dicts in their own insertion order).
  int ix = 0;
  const float* h     = (const float*)d_in[ix++];   // (N,128)
  const float* x     = (const float*)d_in[ix++];   // (N,16,8)
  const int*   edges = (const int*)  d_in[ix++];   // (2,E)
  const float* nah   = (const float*)d_in[ix++];   // (N,2)
  const float* nax   = (const float*)d_in[ix++];   // (N,2,8)
  ix++;                                            // edge_attr_h (E,0) unused
  const float* eax   = (const float*)d_in[ix++];   // (E,3,8)
  const float* px_Wr  = (const float*)d_in[ix++];  // (51,51,4)
  const float* px_Wl  = (const float*)d_in[ix++];  // (16,51,4)
  const float* px_bl  = (const float*)d_in[ix++];  // (16)
  const float* px_Wgp = (const float*)d_in[ix++];  // (16,51,4,4,4)
  const float* px_lna = (const float*)d_in[ix++];  // (16)
  const float* tx_Wr  = (const float*)d_in[ix++];  // (34,34,4)
  const float* tx_Wl  = (const float*)d_in[ix++];  // (16,34,4)
  const float* tx_bl  = (const float*)d_in[ix++];  // (16)
  const float* tx_Wgp = (const float*)d_in[ix++];  // (16,34,4,4,4)
  const float* tx_lna = (const float*)d_in[ix++];  // (16)
  const float* ph_W1  = (const float*)d_in[ix++];  // (128,448)
  const float* ph_g   = (const float*)d_in[ix++];  // (128)
  const float* ph_b   = (const float*)d_in[ix++];  // (128)
  const float* ph_W2  = (const float*)d_in[ix++];  // (128,128)
  const float* ph_b2  = (const float*)d_in[ix++];  // (128)
  const float* th_W1  = (const float*)d_in[ix++];  // (128,322)
  const float* th_b1  = (const float*)d_in[ix++];  // (128)
  const float* th_g   = (const float*)d_in[ix++];  // (128)
  const float* th_b   = (const float*)d_in[ix++];  // (128)
  const float* th_W2  = (const float*)d_in[ix++];  // (128,128)
  const float* th_b2  = (const float*)d_in[ix++];  // (128)
  const float* psi_W1 = (const float*)d_in[ix++];  // (128,128)
  const float* psi_b1 = (const float*)d_in[ix++];  // (128)
  const float* psi_W2 = (const float*)d_in[ix++];  // (64,128)
  const float* psi_b2 = (const float*)d_in[ix++];  // (64)
  const float* chi_W1 = (const float*)d_in[ix++];  // (128,128)
  const float* chi_b1 = (const float*)d_in[ix++];  // (128)
  const float* chi_W2 = (const float*)d_in[ix++];  // (64,128)
  const float* chi_b2 = (const float*)d_in[ix++];  // (64)

  float* out_hu = (float*)d_out;             // (N,128)
  float* out_xu = out_hu + (size_t)NN * 128; // (N,16,8)

  // workspace layout (floats)
  float* w = (float*)d_ws;
  size_t o = 0;
  float* mx    = w + o; o += (size_t)NE * 128;  // m_x (post-LN, gated in place)
  float* minv  = w + o; o += (size_t)NE * 64;
  float* z1    = w + o; o += (size_t)NE * 128;  // reused as psi hidden t
  float* mh    = w + o; o += (size_t)NE * 128;
  float* stats = w + o; o += 1024;              // [0:256) e-sum/sq [256:512) e-mu/rstd
                                                // [512:768) n-sum/sq [768:1024) n-mu/rstd
  float* xacc  = w + o; o += (size_t)NN * 128;  // -> x_red
  float* hacc  = w + o; o += (size_t)NN * 128;  // -> h_red
  float* cnt   = w + o; o += NN;
  float* xupre = w + o; o += (size_t)NN * 128;
  float* uinv  = w + o; o += (size_t)NN * 64;
  float* z2    = w + o; o += (size_t)NN * 128;  // reused as chi hidden t2
  float* w1pad = w + o; o += (size_t)HID * K_TH;
  float* tpsi  = z1;
  float* tchi  = z2;

  // -- init: zero stats + accumulators, pad theta_h_W1 to K=324 --
  zero_kernel<<<(1024 + 255) / 256, 256, 0, stream>>>(stats, 1024);
  zero_kernel<<<((long)NN * 257 + 255) / 256, 256, 0, stream>>>(xacc, (long)NN * 257);
  pad_w1_kernel<<<(HID * K_TH + 255) / 256, 256, 0, stream>>>(th_W1, w1pad);

  // -- edge stage --
  edge_geom_kernel<<<NE / 8, 128, 0, stream>>>(x, edges, eax, px_Wr, px_Wl,
                                               px_bl, px_Wgp, px_lna, mx, minv);
  gemm_phi_h1_kernel<<<NE / 16, 256, 0, stream>>>(minv, h, edges, ph_W1, z1);
  bn_stats_kernel<<<64, 128, 0, stream>>>(z1, NE, stats);
  bn_finalize_kernel<<<1, 128, 0, stream>>>(stats, 1.f / (float)NE, stats + 256);
  gemm_bn_w2_kernel<true><<<NE / 16, 256, 0, stream>>>(z1, stats + 256, ph_g, ph_b,
                                                       ph_W2, ph_b2, mh);
  gemm_relu_kernel<<<NE / 16, 256, 0, stream>>>(mh, psi_W1, psi_b1, tpsi);
  gemm_gate_kernel<<<NE / 16, 128, 0, stream>>>(tpsi, psi_W2, psi_b2, mx, mx);

  // -- segment mean --
  scatter_kernel<<<4096, 256, 0, stream>>>(mx, mh, edges, xacc, hacc, cnt);
  seg_div_kernel<<<(NN * 128 + 255) / 256, 256, 0, stream>>>(xacc, hacc, cnt);

  // -- node stage --
  node_geom_kernel<<<NN / 8, 128, 0, stream>>>(x, xacc, nax, tx_Wr, tx_Wl,
                                               tx_bl, tx_Wgp, tx_lna, xupre, uinv);
  gemm_theta_h1_kernel<<<NN / 16, 256, 0, stream>>>(h, hacc, uinv, nah, w1pad,
                                                    th_b1, z2);
  bn_stats_kernel<<<64, 128, 0, stream>>>(z2, NN, stats + 512);
  bn_finalize_kernel<<<1, 128, 0, stream>>>(stats + 512, 1.f / (float)NN, stats + 768);
  gemm_bn_w2_kernel<false><<<NN / 16, 256, 0, stream>>>(z2, stats + 768, th_g, th_b,
                                                        th_W2, th_b2, out_hu);
  gemm_relu_kernel<<<NN / 16, 256, 0, stream>>>(out_hu, chi_W1, chi_b1, tchi);
  gemm_gate_kernel<<<NN / 16, 128, 0, stream>>>(tchi, chi_W2, chi_b2, xupre, out_xu);
}